// chamfer_distance_loss_64836826300486
// MI455X (gfx1250) — compile-verified
//
#include <hip/hip_runtime.h>
#include <hip/hip_bf16.h>
#include <float.h>

typedef __attribute__((ext_vector_type(2))) float v2f;
typedef __attribute__((ext_vector_type(8))) float v8f;

#define WAVES_PER_BLOCK 8
#define ROWS_PER_WAVE 2   // 16-row tiles per wave

__global__ void chamfer_zero_kernel(float* out) { out[0] = 0.0f; }

// One kernel computes both directions: blockIdx.z==0 -> rows=p1, cols=p2 (dist1)
//                                      blockIdx.z==1 -> rows=p2, cols=p1 (dist2)
// Homogeneous trick: A row fragment a' = (ax,ay,az,||a||^2), B col fragment
// b' = (-2bx,-2by,-2bz,1), and the C accumulator is preloaded with ||b||^2
// (constant per lane since column == lane&15 in the C/D layout). A single
// v_wmma_f32_16x16x4_f32 then yields the full 16x16 squared-distance block;
// the only per-element VALU work left is one v_med3 (clamp-to-0 + running min).
__global__ __launch_bounds__(WAVES_PER_BLOCK * 32)
void chamfer_pass_kernel(const float* __restrict__ p1,
                         const float* __restrict__ p2,
                         float* __restrict__ out,
                         int N, int M) {
  const int lane = threadIdx.x & 31;
  const int wave = threadIdx.x >> 5;
  const int batch = blockIdx.y;
  const int swapRole = blockIdx.z;

  const float* rows = swapRole ? p2 : p1;
  const float* cols = swapRole ? p1 : p2;
  const int nR = swapRole ? M : N;
  const int nC = swapRole ? N : M;

  // contribution scale: per-sample mean over rows, then mean over batch
  const float scale = 1.0f / ((float)gridDim.y * (float)nR);

  const int tileBase = (blockIdx.x * WAVES_PER_BLOCK + wave) * ROWS_PER_WAVE;
  if ((tileBase + ROWS_PER_WAVE) * 16 > nR) return;  // uniform per wave

  const int halfSel = lane >> 4;   // 0: lanes 0-15 (K=0,1), 1: lanes 16-31 (K=2,3)
  const int l15 = lane & 15;

  // ---- Build A fragments: a' = (ax, ay, az, ||a||^2) ----
  // A 16x4 f32 layout: VGPR0 = K=0 (lanes 0-15) / K=2 (lanes 16-31)
  //                    VGPR1 = K=1 (lanes 0-15) / K=3 (lanes 16-31)
  v2f afrag[ROWS_PER_WAVE];
#pragma unroll
  for (int r = 0; r < ROWS_PER_WAVE; ++r) {
    const int m = (tileBase + r) * 16 + l15;
    const float* ap = rows + ((size_t)batch * nR + m) * 3;
    const float ax = ap[0], ay = ap[1], az = ap[2];
    const float aa = ax * ax + ay * ay + az * az;
    afrag[r].x = halfSel ? az : ax;  // K=2 : K=0
    afrag[r].y = halfSel ? aa : ay;  // K=3 : K=1
  }

  float runmin[ROWS_PER_WAVE][8];
#pragma unroll
  for (int r = 0; r < ROWS_PER_WAVE; ++r)
#pragma unroll
    for (int v = 0; v < 8; ++v) runmin[r][v] = FLT_MAX;

  const float* cbase = cols + (size_t)batch * nC * 3;
  const int nTiles = nC >> 4;

  // Software-pipelined load: fetch tile t+1 while computing tile t.
  float nbx, nby, nbz;
  {
    const float* bp0 = cbase + (size_t)l15 * 3;
    nbx = bp0[0]; nby = bp0[1]; nbz = bp0[2];
  }

  for (int t = 0; t < nTiles; ++t) {
    const float bx = nbx, by = nby, bz = nbz;
    if (t + 1 < nTiles) {
      const float* bp = cbase + ((size_t)((t + 1) * 16 + l15)) * 3;
      __builtin_prefetch(bp + 48, 0, 3);  // tile t+2 into near cache
      nbx = bp[0]; nby = bp[1]; nbz = bp[2];
    }

    const float bb = bx * bx + by * by + bz * bz;

    // ---- B fragment: b' = (-2bx, -2by, -2bz, 1) ----
    v2f bfrag;
    bfrag.x = halfSel ? (-2.0f * bz) : (-2.0f * bx);  // K=2 : K=0
    bfrag.y = halfSel ? 1.0f : (-2.0f * by);          // K=3 : K=1

    // ---- C preloaded with ||b||^2 (column-constant per lane) ----
    v8f cin;
#pragma unroll
    for (int v = 0; v < 8; ++v) cin[v] = bb;

#pragma unroll
    for (int r = 0; r < ROWS_PER_WAVE; ++r) {
      // D = A x B + C : full 16x16 squared-distance block
      const v8f d = __builtin_amdgcn_wmma_f32_16x16x4_f32(
          /*neg_a=*/false, afrag[r], /*neg_b=*/false, bfrag,
          /*c_mod=*/(short)0, cin, /*reuse_a=*/false, /*reuse_b=*/false);
#pragma unroll
      for (int v = 0; v < 8; ++v) {
        // runmin = min(runmin, max(d, 0)) ; runmin >= 0 so med3 is exact
        runmin[r][v] = __builtin_amdgcn_fmed3f(0.0f, d[v], runmin[r][v]);
      }
    }
  }

  // ---- Reduce: min over the 16 lanes of each half (columns), sum rows ----
  float s = 0.0f;
#pragma unroll
  for (int r = 0; r < ROWS_PER_WAVE; ++r) {
#pragma unroll
    for (int v = 0; v < 8; ++v) {
      float val = runmin[r][v];
      val = fminf(val, __shfl_xor(val, 1, 32));
      val = fminf(val, __shfl_xor(val, 2, 32));
      val = fminf(val, __shfl_xor(val, 4, 32));
      val = fminf(val, __shfl_xor(val, 8, 32));
      s += val;  // lanes 0-15: rows v; lanes 16-31: rows v+8
    }
  }
  s += __shfl_xor(s, 16, 32);  // combine both halves
  s *= scale;
  if (lane == 0) atomicAdd(out, s);  // global_atomic_add_f32
}

extern "C" void kernel_launch(void* const* d_in, const int* in_sizes, int n_in,
                              void* d_out, int out_size, void* d_ws, size_t ws_size,
                              hipStream_t stream) {
  (void)n_in; (void)out_size; (void)d_ws; (void)ws_size;
  const float* p1 = (const float*)d_in[0];
  const float* p2 = (const float*)d_in[1];
  float* out = (float*)d_out;

  const int B = 4;
  const int N = in_sizes[0] / (B * 3);
  const int M = in_sizes[1] / (B * 3);

  chamfer_zero_kernel<<<1, 1, 0, stream>>>(out);

  const int rowsPerBlock = 16 * ROWS_PER_WAVE * WAVES_PER_BLOCK;  // 256
  const int maxR = (N > M) ? N : M;
  dim3 block(WAVES_PER_BLOCK * 32);
  dim3 grid((maxR + rowsPerBlock - 1) / rowsPerBlock, B, 2);
  chamfer_pass_kernel<<<grid, block, 0, stream>>>(p1, p2, out, N, M);
}